// FlowAlignedBilateralEffect_5849745457655
// MI455X (gfx1250) — compile-verified
//
#include <hip/hip_runtime.h>
#include <math.h>

// ---------------- problem constants (fixed by setup_inputs) ----------------
constexpr int W  = 1024;
constexpr int H  = 768;
constexpr int CH = 3;
constexpr int BS = 2;

constexpr int TW   = 32;           // output tile width
constexpr int TH   = 32;           // output tile height
constexpr int HALO = 12;           // max displacement is <= 11 px (10 + bilinear)
constexpr int LW   = TW + 2 * HALO;   // 56
constexpr int LH   = TH + 2 * HALO;   // 56
constexpr int LSTRIDE = LW + 1;       // 57 float4 slots per row (pad vs bank conflicts)

constexpr float FEPS = 1.1920928955078125e-07f;  // np.finfo(float32).eps

#define LDS_AS __attribute__((address_space(3)))
#define GLB_AS __attribute__((address_space(1)))

typedef GLB_AS int* as1_int_ptr;
typedef LDS_AS int* as3_int_ptr;

#if defined(__gfx1250__) && __has_builtin(__builtin_amdgcn_global_load_async_to_lds_b32)
#define HAVE_ASYNC_LDS 1
#else
#define HAVE_ASYNC_LDS 0
#endif

__device__ __forceinline__ int clampi(int v, int lo, int hi) {
    return v < lo ? lo : (v > hi ? hi : v);
}

__device__ __forceinline__ float4 ld4(const float* tile, int ly, int lx) {
    return *(const float4*)&tile[(ly * LSTRIDE + lx) * 4];
}

// Border-clamped bilinear sample from the LDS tile (RGB in float4 slots).
__device__ __forceinline__ float4 bsample(const float* tile, float xp, float yp,
                                          int tx0, int ty0) {
    float xf = floorf(xp), yf = floorf(yp);
    float fx = xp - xf,    fy = yp - yf;
    int x0 = (int)xf, y0 = (int)yf;
    int x0c = clampi(x0,     0, W - 1);
    int x1c = clampi(x0 + 1, 0, W - 1);
    int y0c = clampi(y0,     0, H - 1);
    int y1c = clampi(y0 + 1, 0, H - 1);
    // map image coords -> LDS tile coords (guaranteed in [0, L?-1] by halo bound)
    int lx0 = x0c - tx0 + HALO, lx1 = x1c - tx0 + HALO;
    int ly0 = y0c - ty0 + HALO, ly1 = y1c - ty0 + HALO;
    float4 v00 = ld4(tile, ly0, lx0);
    float4 v01 = ld4(tile, ly0, lx1);
    float4 v10 = ld4(tile, ly1, lx0);
    float4 v11 = ld4(tile, ly1, lx1);
    float gx = 1.0f - fx, gy = 1.0f - fy;
    float4 r;
    r.x = (v00.x * gx + v01.x * fx) * gy + (v10.x * gx + v11.x * fx) * fy;
    r.y = (v00.y * gx + v01.y * fx) * gy + (v10.y * gx + v11.y * fx) * fy;
    r.z = (v00.z * gx + v01.z * fx) * gy + (v10.z * gx + v11.z * fx) * fy;
    r.w = 0.0f;
    return r;
}

__global__ __launch_bounds__(256) void fab_kernel(
    const float* __restrict__ x, const float* __restrict__ tangent,
    const float* __restrict__ sigmaD, const float* __restrict__ sigmaR,
    float* __restrict__ out)
{
    __shared__ __align__(16) float tile[LH * LSTRIDE * 4];   // ~51 KB

    const int b   = blockIdx.z;
    const int tx0 = blockIdx.x * TW;
    const int ty0 = blockIdx.y * TH;
    const int tid = threadIdx.x;

    const size_t hw = (size_t)H * W;
    const float* xb = x + (size_t)b * CH * hw;

    // ---- stage x tile (+halo, border-clamped) into LDS, RGB interleaved ----
    for (int p = tid; p < LH * LW; p += 256) {
        int ly = p / LW, lx = p - ly * LW;
        int gy = clampi(ty0 - HALO + ly, 0, H - 1);
        int gx = clampi(tx0 - HALO + lx, 0, W - 1);
        int lidx = (ly * LSTRIDE + lx) * 4;
        const float* src = xb + (size_t)gy * W + gx;
#if HAVE_ASYNC_LDS
        __builtin_amdgcn_global_load_async_to_lds_b32(
            (as1_int_ptr)(src),          (as3_int_ptr)&tile[lidx + 0], 0, 0);
        __builtin_amdgcn_global_load_async_to_lds_b32(
            (as1_int_ptr)(src + hw),     (as3_int_ptr)&tile[lidx + 1], 0, 0);
        __builtin_amdgcn_global_load_async_to_lds_b32(
            (as1_int_ptr)(src + 2 * hw), (as3_int_ptr)&tile[lidx + 2], 0, 0);
#else
        tile[lidx + 0] = src[0];
        tile[lidx + 1] = src[hw];
        tile[lidx + 2] = src[2 * hw];
#endif
    }
#if HAVE_ASYNC_LDS
#if __has_builtin(__builtin_amdgcn_s_wait_asynccnt)
    __builtin_amdgcn_s_wait_asynccnt(0);
#else
    asm volatile("s_wait_asynccnt 0" ::: "memory");
#endif
#endif
    __syncthreads();

    // ---- per-batch scalars ----
    const float sD = sigmaD[b];
    const float sR = sigmaR[b];
    const float inv2sD2   = 1.0f / (2.0f * sD * sD + FEPS);
    const float inv2sR2   = 1.0f / (2.0f * sR * sR + FEPS);
    const float half_step = 2.0f * sD;

    const int col  = tid & (TW - 1);
    const int row0 = tid >> 5;                  // 0..7, 4 rows per thread
    const float* tgx = tangent + (size_t)b * 2 * hw;
    const float* tgy = tgx + hw;
    float* outb = out + (size_t)b * CH * hw;

    for (int k = 0; k < 4; ++k) {
        const int row = row0 + k * 8;
        const int px = tx0 + col;
        const int py = ty0 + row;
        const size_t pix = (size_t)py * W + px;

        const float tvx = tgx[pix];
        const float tvy = tgy[pix];
        const float me  = fmaxf(fabsf(tvx), fabsf(tvy));
        const float ds  = (me > 0.0f) ? (1.0f / me) : 1.0f;

        const float4 ctr = ld4(tile, row + HALO, col + HALO);

        float nrm = 0.0f, aR = 0.0f, aG = 0.0f, aB = 0.0f;

#pragma unroll
        for (int t = 0; t < 5; ++t) {
            const float i1 = ds * (float)(2 * t + 1);
            // masked taps (|itr1| > half_step) have kernelD == 0 in the reference
            // and contribute exactly nothing -> skip (also avoids itr_d/eps blowup)
            if (!(i1 <= half_step)) continue;
            const float i2 = ds * (float)(2 * t + 2);
            const float kD1 = __expf(-(i1 * i1) * inv2sD2);
            const float kD2 = __expf(-(i2 * i2) * inv2sD2);
            const float kD  = kD1 + kD2;
            const float D   = (i1 * kD1 + i2 * kD2) / (kD + FEPS);
            const float dx = D * tvx, dy = D * tvy;     // displacement in pixels

            const float4 c0 = bsample(tile, (float)px + dx, (float)py + dy, tx0, ty0);
            const float4 c1 = bsample(tile, (float)px - dx, (float)py - dy, tx0, ty0);

            float dr = (c0.x - ctr.x) * 100.0f;
            float dg = (c0.y - ctr.y) * 254.0f;
            float db = (c0.z - ctr.z) * 254.0f;
            const float e0 = sqrtf(fmaxf(dr * dr + dg * dg + db * db, 1e-30f));
            dr = (c1.x - ctr.x) * 100.0f;
            dg = (c1.y - ctr.y) * 254.0f;
            db = (c1.z - ctr.z) * 254.0f;
            const float e1 = sqrtf(fmaxf(dr * dr + dg * dg + db * db, 1e-30f));

            const float w0 = kD * __expf(-(e0 * e0) * inv2sR2);
            const float w1 = kD * __expf(-(e1 * e1) * inv2sR2);

            nrm += w0 + w1;
            aR += w0 * c0.x + w1 * c1.x;
            aG += w0 * c0.y + w1 * c1.y;
            aB += w0 * c0.z + w1 * c1.z;
        }

        const float inv = 1.0f / (1.0f + nrm);
        outb[pix]          = (ctr.x + aR) * inv;
        outb[hw + pix]     = (ctr.y + aG) * inv;
        outb[2 * hw + pix] = (ctr.z + aB) * inv;
    }
}

extern "C" void kernel_launch(void* const* d_in, const int* in_sizes, int n_in,
                              void* d_out, int out_size, void* d_ws, size_t ws_size,
                              hipStream_t stream) {
    const float* x       = (const float*)d_in[0];
    const float* tangent = (const float*)d_in[1];
    const float* sigmaD  = (const float*)d_in[2];
    const float* sigmaR  = (const float*)d_in[3];
    float* out = (float*)d_out;

    dim3 grid(W / TW, H / TH, BS);   // 32 x 24 x 2
    fab_kernel<<<grid, 256, 0, stream>>>(x, tangent, sigmaD, sigmaR, out);
}